// HardNegSupConLoss_65120294142604
// MI455X (gfx1250) — compile-verified
//
#include <hip/hip_runtime.h>
#include <hip/hip_bf16.h>
#include <math.h>

// ---------------------------------------------------------------------------
// HardNegSupConLoss for MI455X (gfx1250, wave32)
//   B = 8192 samples, D = 128 dims, 100 classes, scalar f32 output.
//
//   Stage 1: label histogram -> counts[100]; k = max(1, int(0.5*(B - sum c^2/B)))
//   Stage 2: (per 1024-row chunk, 8 chunks; 32MB chunk stays L2-resident)
//       2a: sim[chunkRows, :] = z @ z^T  via V_WMMA_F32_16X16X4_F32,
//           B operand staged through LDS (conflict-free padded layout) so the
//           8 waves of a workgroup don't re-load identical global data
//       2b: one workgroup per row: stage row in LDS, exact radix-select of the
//           k-th largest negative sim (threshold), masked softmax loss terms
//   Stage 3: deterministic tree reduction -> scalar loss
// ---------------------------------------------------------------------------

typedef __attribute__((ext_vector_type(2))) float v2f;
typedef __attribute__((ext_vector_type(8))) float v8f;

#define NB        8192
#define ND        128
#define NCLS      100
#define CHUNK     1024
#define NCHUNK    (NB / CHUNK)
#define TEMP_INV  (1.0f / 0.07f)
#define NEG_FILL  (-1.0e9f)

#define KSLAB     64
#define BPAD      68   // 64 + 4 floats pad: per-column bank advance = 4 -> conflict-free

// ---------------- Stage 1: label histogram + k ------------------------------

__global__ void hnc_zero_counts(int* counts) {
    int t = threadIdx.x;
    if (t < NCLS) counts[t] = 0;
}

__global__ void hnc_label_hist(const int* __restrict__ labels, int* counts) {
    int i = blockIdx.x * blockDim.x + threadIdx.x;
    if (i < NB) atomicAdd(&counts[labels[i]], 1);
}

__global__ void hnc_compute_k(const int* __restrict__ counts, int* kOut) {
    __shared__ long long sh[128];
    int tid = threadIdx.x;
    long long s = 0;
    for (int c = tid; c < NCLS; c += 128) {
        long long cc = counts[c];
        s += cc * cc;
    }
    sh[tid] = s;
    __syncthreads();
    for (int st = 64; st > 0; st >>= 1) {
        if (tid < st) sh[tid] += sh[tid + st];
        __syncthreads();
    }
    if (tid == 0) {
        long long sumsq = sh[0];
        long long num   = (long long)NB * (long long)NB - sumsq; // sum of per-row #neg
        float mean = (float)num / (float)NB;        // mimic f32 mean
        int  k = (int)((double)mean * 0.5);         // python float()*0.5, int() truncates
        if (k < 1) k = 1;
        *kOut = k;
    }
}

// ---------------- Stage 2a: sim chunk GEMM via WMMA f32 ---------------------
// Workgroup = 256 threads (8 waves). Tile = 128 rows x 128 cols.
// Wave w owns rows [tile + 16w, +16), 8 column sub-tiles of 16.
// A(16x4) lane layout: lane<16 -> {z[r][k], z[r][k+1]}, lane>=16 -> k+2,k+3
// (wave-private rows, loaded straight from global: no redundancy).
// B(4x16) mirrors it on the column index; the B slab (128 cols x 64 k) is
// shared by all 8 waves, so it is staged once per workgroup into LDS with a
// 68-float column stride: lanes 0..15 hit banks {4c}, lanes 16..31 {4c+2},
// each ds_load_b64 covers a bank pair -> all 64 banks used exactly once.

__global__ __launch_bounds__(256) void hnc_gemm(const float* __restrict__ z,
                                                float* __restrict__ simChunk,
                                                int rowBase) {
    __shared__ float sB[128 * BPAD];   // ~34 KB

    const int wave = threadIdx.x >> 5;
    const int lane = threadIdx.x & 31;
    const int lmod = lane & 15;
    const int lhi  = lane >> 4;            // 0 or 1 -> K pair select

    const int mBlock = blockIdx.y;         // 0..CHUNK/128-1
    const int nBlock = blockIdx.x;         // 0..NB/128-1
    const int rowG   = rowBase + mBlock * 128 + wave * 16;  // global z row for A
    const int colG   = nBlock * 128;                        // global z row for B

    const float* __restrict__ aRow = z + (size_t)(rowG + lmod) * ND;

    v8f acc[8];
#pragma unroll
    for (int t = 0; t < 8; ++t) acc[t] = v8f{0.f,0.f,0.f,0.f,0.f,0.f,0.f,0.f};

    for (int k0 = 0; k0 < ND; k0 += KSLAB) {
        __syncthreads();   // previous slab fully consumed
        // cooperative stage: 128 cols x 64 k = 32KB, b128 per thread per step,
        // 16 threads cover one column (256B) -> fully coalesced, conflict-free
        for (int idx = (int)threadIdx.x; idx < 128 * (KSLAB / 4); idx += 256) {
            const int c = idx >> 4;
            const int q = idx & 15;
            const float4 v = *(const float4*)(z + (size_t)(colG + c) * ND + k0 + q * 4);
            *(float4*)(&sB[c * BPAD + q * 4]) = v;
        }
        __syncthreads();

#pragma unroll 4
        for (int kk = 0; kk < KSLAB; kk += 4) {
            v2f a = *(const v2f*)(aRow + k0 + kk + 2 * lhi);
#pragma unroll
            for (int t = 0; t < 8; ++t) {
                v2f b = *(const v2f*)(&sB[(t * 16 + lmod) * BPAD + kk + 2 * lhi]);
                acc[t] = __builtin_amdgcn_wmma_f32_16x16x4_f32(
                    false, a, false, b, (short)0, acc[t], false, false);
            }
        }
    }

    // C/D layout: VGPR v: lanes 0-15 -> (M=v, N=lane); lanes 16-31 -> (M=v+8, N=lane-16)
    const int simRow0 = mBlock * 128 + wave * 16;  // chunk-relative
#pragma unroll
    for (int t = 0; t < 8; ++t) {
        const int c = colG + t * 16 + lmod;
#pragma unroll
        for (int v = 0; v < 8; ++v) {
            const int r = simRow0 + v + 8 * lhi;
            simChunk[(size_t)r * NB + c] = acc[t][v];
        }
    }
}

// ---------------- Stage 2b: per-row select + loss ---------------------------

__device__ __forceinline__ unsigned orderKey(float f) {
    unsigned u = __float_as_uint(f);
    return (u & 0x80000000u) ? ~u : (u | 0x80000000u);   // monotone: bigger key = bigger float
}
__device__ __forceinline__ float keyToFloat(unsigned m) {
    unsigned u = (m & 0x80000000u) ? (m & 0x7FFFFFFFu) : ~m;
    return __uint_as_float(u);
}

__device__ __forceinline__ int blockReduceSumI(int v, int* buf) {
    int tid = threadIdx.x;
    buf[tid] = v; __syncthreads();
    for (int s = 128; s > 0; s >>= 1) {
        if (tid < s) buf[tid] += buf[tid + s];
        __syncthreads();
    }
    int r = buf[0]; __syncthreads();
    return r;
}
__device__ __forceinline__ float blockReduceSumF(float v, float* buf) {
    int tid = threadIdx.x;
    buf[tid] = v; __syncthreads();
    for (int s = 128; s > 0; s >>= 1) {
        if (tid < s) buf[tid] += buf[tid + s];
        __syncthreads();
    }
    float r = buf[0]; __syncthreads();
    return r;
}
__device__ __forceinline__ float blockReduceMaxF(float v, float* buf) {
    int tid = threadIdx.x;
    buf[tid] = v; __syncthreads();
    for (int s = 128; s > 0; s >>= 1) {
        if (tid < s) buf[tid] = fmaxf(buf[tid], buf[tid + s]);
        __syncthreads();
    }
    float r = buf[0]; __syncthreads();
    return r;
}

__global__ __launch_bounds__(256) void hnc_rowproc(const float* __restrict__ simChunk,
                                                   const int* __restrict__ labels,
                                                   const int* __restrict__ kPtr,
                                                   float* __restrict__ rowLoss,
                                                   int* __restrict__ rowNpos,
                                                   int rowBase) {
    __shared__ float         sSim[NB];      // 32KB: full sim row
    __shared__ unsigned char sMask[NB];     // 8KB:  0=neg 1=pos 2=diag
    __shared__ unsigned int  hist[256];     // radix histogram
    __shared__ float         fRed[256];
    __shared__ int           iRed[256];
    __shared__ int           bcast[2];

    const int tid  = threadIdx.x;
    const int row  = rowBase + blockIdx.x;            // global row id
    const int myLab = labels[row];
    const float* __restrict__ simRow = simChunk + (size_t)blockIdx.x * NB;

    // --- stage row + masks, count pos/neg ---
    int nPosL = 0, nNegL = 0;
    for (int j = tid; j < NB; j += 256) {
        float s = simRow[j];
        sSim[j] = s;
        unsigned char m = (j == row) ? 2 : ((labels[j] == myLab) ? 1 : 0);
        sMask[j] = m;
        nPosL += (m == 1);
        nNegL += (m == 0);
    }
    __syncthreads();
    const int nPos = blockReduceSumI(nPosL, iRed);
    const int nNeg = blockReduceSumI(nNegL, iRed);
    const int kVal = *kPtr;

    // --- exact k-th largest negative sim via 4x8-bit MSB radix select ---
    float threshF;
    if (kVal > nNeg) {
        threshF = NEG_FILL;      // top_k hit the -1e9 fill -> all negatives are "hard"
    } else {
        unsigned prefix = 0, pmask = 0;
        int kRem = kVal;
        for (int shift = 24; shift >= 0; shift -= 8) {
            hist[tid] = 0;
            __syncthreads();
            for (int j = tid; j < NB; j += 256) {
                if (sMask[j] == 0) {
                    unsigned u = orderKey(sSim[j]);
                    if ((u & pmask) == prefix)
                        atomicAdd(&hist[(u >> shift) & 0xFFu], 1u);
                }
            }
            __syncthreads();
            if (tid == 0) {
                int acc = 0, b = 255;
                for (; b > 0; --b) {
                    int c = (int)hist[b];
                    if (acc + c >= kRem) break;
                    acc += c;
                }
                bcast[0] = b;
                bcast[1] = kRem - acc;
            }
            __syncthreads();
            prefix |= ((unsigned)bcast[0]) << shift;
            pmask  |= 0xFFu << shift;
            kRem    = bcast[1];
            __syncthreads();
        }
        threshF = keyToFloat(prefix);   // exact element value of k-th largest negative
    }

    // --- row max over active = pos | (neg & sim >= thresh) ---
    float mL = -3.4e38f;
    for (int j = tid; j < NB; j += 256) {
        unsigned char m = sMask[j];
        float s = sSim[j];
        bool active = (m == 1) || (m == 0 && s >= threshF);
        if (active) mL = fmaxf(mL, s);
    }
    float mAct = blockReduceMaxF(mL, fRed);
    bool anyActive = (mAct > -3.0e38f);
    float rm = anyActive ? mAct * TEMP_INV : NEG_FILL;  // row_max on ss scale

    // --- denom / positive term sums ---
    float denL = 0.f, posL = 0.f;
    for (int j = tid; j < NB; j += 256) {
        unsigned char m = sMask[j];
        float s = sSim[j];
        bool active = (m == 1) || (m == 0 && s >= threshF);
        if (active) {
            float ssj = s * TEMP_INV - rm;
            denL += expf(ssj);
            if (m == 1) posL += ssj;
        }
    }
    float den    = blockReduceSumF(denL, fRed);
    float posSum = blockReduceSumF(posL, fRed);

    if (tid == 0) {
        float logden = logf(den + 1e-9f);
        float np = (float)nPos;
        float loss = -(posSum - np * logden) / fmaxf(np, 1.0f);
        rowLoss[row] = loss;
        rowNpos[row] = nPos;
    }
}

// ---------------- Stage 3: deterministic final reduction --------------------

__global__ void hnc_finalize(const float* __restrict__ rowLoss,
                             const int* __restrict__ rowNpos,
                             float* __restrict__ out) {
    __shared__ float fb[256];
    __shared__ int   ib[256];
    int tid = threadIdx.x;
    float ls = 0.f;
    int hs = 0;
    for (int i = tid; i < NB; i += 256) {
        if (rowNpos[i] > 0) { ls += rowLoss[i]; hs++; }
    }
    fb[tid] = ls; ib[tid] = hs;
    __syncthreads();
    for (int s = 128; s > 0; s >>= 1) {
        if (tid < s) { fb[tid] += fb[tid + s]; ib[tid] += ib[tid + s]; }
        __syncthreads();
    }
    if (tid == 0) {
        // if no row has positives every loss is 0, so the loss.mean() branch is 0 too
        out[0] = (ib[0] > 0) ? fb[0] / (float)ib[0] : 0.0f;
    }
}

// ---------------------------------------------------------------------------

extern "C" void kernel_launch(void* const* d_in, const int* in_sizes, int n_in,
                              void* d_out, int out_size, void* d_ws, size_t ws_size,
                              hipStream_t stream) {
    const float* z      = (const float*)d_in[0];
    const int*   labels = (const int*)d_in[1];
    float*       out    = (float*)d_out;

    char* ws = (char*)d_ws;
    float* simChunk = (float*)ws;                                   // 32 MB (L2-resident)
    const size_t SIM_BYTES = (size_t)CHUNK * NB * sizeof(float);
    int*   counts  = (int*)(ws + SIM_BYTES);                        // 100 (padded to 128)
    int*   kPtr    = counts + 128;
    float* rowLoss = (float*)(kPtr + 32);                           // 8192 f32
    int*   rowNpos = (int*)(rowLoss + NB);                          // 8192 i32

    hnc_zero_counts<<<1, 128, 0, stream>>>(counts);
    hnc_label_hist<<<NB / 256, 256, 0, stream>>>(labels, counts);
    hnc_compute_k<<<1, 128, 0, stream>>>(counts, kPtr);

    for (int c = 0; c < NCHUNK; ++c) {
        const int rowBase = c * CHUNK;
        dim3 gemmGrid(NB / 128, CHUNK / 128);
        hnc_gemm<<<gemmGrid, 256, 0, stream>>>(z, simChunk, rowBase);
        hnc_rowproc<<<CHUNK, 256, 0, stream>>>(simChunk, labels, kPtr,
                                               rowLoss, rowNpos, rowBase);
    }

    hnc_finalize<<<1, 256, 0, stream>>>(rowLoss, rowNpos, out);
}